// InfiniAttention_70884140253414
// MI455X (gfx1250) — compile-verified
//
#include <hip/hip_runtime.h>
#include <math.h>

typedef __attribute__((ext_vector_type(2))) float v2f;
typedef __attribute__((ext_vector_type(8))) float v8f;

#define WMMA_F32(a, b, c) \
    __builtin_amdgcn_wmma_f32_16x16x4_f32(false, (a), false, (b), (short)0, (c), false, false)

// gfx1250 async global->LDS path, guarded so the file always compiles.
#if defined(__has_builtin)
#if __has_builtin(__builtin_amdgcn_global_load_async_to_lds_b128) && \
    __has_builtin(__builtin_amdgcn_s_wait_asynccnt)
#define USE_ASYNC 1
#endif
#endif
#ifndef USE_ASYNC
#define USE_ASYNC 0
#endif

#if USE_ASYNC
// Exact pointee type from the compiler diagnostic:
//   "__attribute__((__vector_size__(4 * sizeof(int)))) int __device__ *"
typedef int v4i_mem __attribute__((vector_size(4 * sizeof(int))));
typedef __attribute__((address_space(1))) v4i_mem* gas_v4i;   // global (AS1)
typedef __attribute__((address_space(3))) v4i_mem* las_v4i;   // LDS (AS3)
#endif

// Copy 16 aligned bytes global -> LDS. Async when available (ASYNCcnt-tracked),
// else synchronous float4 via VGPRs.
__device__ __forceinline__ void async_copy16(const float* __restrict__ g,
                                             float* __restrict__ l) {
#if USE_ASYNC
    __builtin_amdgcn_global_load_async_to_lds_b128(
        (gas_v4i)(unsigned long long)g,
        (las_v4i)(unsigned)(unsigned long long)l,
        0, 0);
#else
    float4 f = *(const float4*)g;
    l[0] = f.x; l[1] = f.y; l[2] = f.z; l[3] = f.w;
#endif
}

__device__ __forceinline__ void wait_async_le(int pending_next) {
#if USE_ASYNC
    if (pending_next > 0) __builtin_amdgcn_s_wait_asynccnt(12);  // covers 4 or 12
    else                  __builtin_amdgcn_s_wait_asynccnt(0);
#else
    (void)pending_next;
#endif
}

__device__ __forceinline__ float sigma_elu1(float x) {
    // elu(x) + 1 : strictly positive feature map
    return x > 0.0f ? x + 1.0f : __expf(x);
}

// ---------------------------------------------------------------------------
// Big GEMM: Y = X (Mdim x Kdim) @ W^T (W is Ndim x Kdim, row-major)
// mode 0: Y[m*Ndim + n]
// mode 1: head-split scatter Y[(((b*16+h)*4096)+t)*64 + d]
// 128x128 WG tile, 8 waves, wave tile 32x64 (2x4 WMMA 16x16 tiles),
// double-buffered K tiles streamed with async global->LDS copies.
// Row stride 20 floats = 80B: 16B-aligned rows for B128 async stores and
// conflict-free column reads (gcd(20,64)=4 -> 16 distinct banks).
// ---------------------------------------------------------------------------
__global__ __launch_bounds__(256) void gemm_xwT(const float* __restrict__ X,
                                                const float* __restrict__ W,
                                                float* __restrict__ Y,
                                                int Mdim, int Ndim, int Kdim,
                                                int mode) {
    __shared__ float As[2][128][20];
    __shared__ float Bs[2][128][20];

    const int tid  = threadIdx.x;
    const int lane = tid & 31;
    const int wave = tid >> 5;
    const int half = lane >> 4;
    const int r    = lane & 15;
    const int m0 = blockIdx.y * 128;
    const int n0 = blockIdx.x * 128;
    const int wm = (wave >> 1) * 32;
    const int wn = (wave & 1) * 64;

    v8f acc[2][4];
#pragma unroll
    for (int mi = 0; mi < 2; ++mi)
#pragma unroll
        for (int ni = 0; ni < 4; ++ni) acc[mi][ni] = {};

    // issue one 128x16 K-tile of X and W into buffer `bf` (4 async ops/thread)
    auto issue = [&](int bf, int k0) {
        for (int i = tid; i < 512; i += 256) {
            const int row = i >> 2;
            const int kc  = (i & 3) * 4;
            async_copy16(&X[(size_t)(m0 + row) * Kdim + k0 + kc], &As[bf][row][kc]);
            async_copy16(&W[(size_t)(n0 + row) * Kdim + k0 + kc], &Bs[bf][row][kc]);
        }
    };

    issue(0, 0);
    int buf = 0;
    for (int k0 = 0; k0 < Kdim; k0 += 16) {
        const bool has_next = (k0 + 16) < Kdim;
        if (has_next) issue(buf ^ 1, k0 + 16);
        wait_async_le(has_next ? 4 : 0);
        __syncthreads();

#pragma unroll
        for (int kb = 0; kb < 4; ++kb) {
            const int k = kb * 4 + 2 * half;
            v2f aF[2], bF[4];
#pragma unroll
            for (int mi = 0; mi < 2; ++mi) {
                const int row = wm + mi * 16 + r;
                aF[mi][0] = As[buf][row][k];
                aF[mi][1] = As[buf][row][k + 1];
            }
#pragma unroll
            for (int ni = 0; ni < 4; ++ni) {
                const int col = wn + ni * 16 + r;
                bF[ni][0] = Bs[buf][col][k];
                bF[ni][1] = Bs[buf][col][k + 1];
            }
#pragma unroll
            for (int mi = 0; mi < 2; ++mi)
#pragma unroll
                for (int ni = 0; ni < 4; ++ni)
                    acc[mi][ni] = WMMA_F32(aF[mi], bF[ni], acc[mi][ni]);
        }
        __syncthreads();   // all waves done with `buf` before it is re-filled
        buf ^= 1;
    }

#pragma unroll
    for (int mi = 0; mi < 2; ++mi)
#pragma unroll
        for (int ni = 0; ni < 4; ++ni)
#pragma unroll
            for (int r8 = 0; r8 < 8; ++r8) {
                const int gm = m0 + wm + mi * 16 + r8 + 8 * half;
                const int gn = n0 + wn + ni * 16 + r;
                const float val = acc[mi][ni][r8];
                if (mode == 0) {
                    Y[(size_t)gm * Ndim + gn] = val;
                } else {
                    const int b = gm >> 12, t = gm & 4095;
                    const int h = gn >> 6,  d = gn & 63;
                    Y[((((size_t)b * 16 + h) << 12) + t) * 64 + d] = val;
                }
            }
}

// ---------------------------------------------------------------------------
// 64x64x64 GEMM on LDS tiles via WMMA (8 waves, 2 tiles each).
// A(m,k) = TRA ? A[k][m] : A[m][k]  (SIGA applies elu+1 on the fly)
// B(k,n) = TRB ? B[n][k] : B[k][n];  C = alpha*A@B (+C if ACC)
// Stride 68 floats: 16B-aligned rows, conflict-free columns (68 % 64 == 4).
// ---------------------------------------------------------------------------
template <bool TRA, bool TRB, bool SIGA, bool ACC>
__device__ __forceinline__ void gemm64(const float (*A)[68], const float (*B)[68],
                                       float (*C)[68], float alpha, int tid) {
    const int lane = tid & 31;
    const int wave = tid >> 5;
    const int half = lane >> 4;
    const int r    = lane & 15;
    const int mi   = wave >> 1;
    const int n0   = (wave & 1) * 32;
    const int mrow = mi * 16 + r;

    v8f acc0 = {}, acc1 = {};
#pragma unroll
    for (int kk = 0; kk < 64; kk += 4) {
        const int k = kk + 2 * half;
        float a0 = TRA ? A[k][mrow]     : A[mrow][k];
        float a1 = TRA ? A[k + 1][mrow] : A[mrow][k + 1];
        if (SIGA) { a0 = sigma_elu1(a0); a1 = sigma_elu1(a1); }
        v2f a; a[0] = a0; a[1] = a1;
        v2f b0, b1;
        b0[0] = TRB ? B[n0 + r][k]          : B[k][n0 + r];
        b0[1] = TRB ? B[n0 + r][k + 1]      : B[k + 1][n0 + r];
        b1[0] = TRB ? B[n0 + 16 + r][k]     : B[k][n0 + 16 + r];
        b1[1] = TRB ? B[n0 + 16 + r][k + 1] : B[k + 1][n0 + 16 + r];
        acc0 = WMMA_F32(a, b0, acc0);
        acc1 = WMMA_F32(a, b1, acc1);
    }
#pragma unroll
    for (int r8 = 0; r8 < 8; ++r8) {
        const int row = mi * 16 + r8 + 8 * half;
        float v0 = alpha * acc0[r8];
        float v1 = alpha * acc1[r8];
        if (ACC) { v0 += C[row][n0 + r]; v1 += C[row][n0 + 16 + r]; }
        C[row][n0 + r]      = v0;
        C[row][n0 + 16 + r] = v1;
    }
}

// ---------------------------------------------------------------------------
// Infini-attention scan: one workgroup per (b,h); M (64x64) and z (64) carried
// in LDS across 64 sequential segments. Next segment's q/k/v tiles are
// prefetched with async global->LDS copies while the current one computes.
// ---------------------------------------------------------------------------
__global__ __launch_bounds__(256) void infini_attn(
    const float* __restrict__ Q, const float* __restrict__ K,
    const float* __restrict__ V, const float* __restrict__ beta_param,
    float* __restrict__ Aout, float* __restrict__ Mws, float* __restrict__ zws) {
    __shared__ float Ml[64][68];
    __shared__ float q2[2][64][68];
    __shared__ float k2[2][64][68];
    __shared__ float v2[2][64][68];
    __shared__ float T1[64][68];
    __shared__ float T2[64][68];
    __shared__ float zl[64];
    __shared__ float dn[64];

    const int tid = threadIdx.x;
    const int bh  = blockIdx.x;
    const int b   = bh >> 4;
    const int h   = bh & 15;
    const float betaH = 1.0f / (1.0f + __expf(-beta_param[h]));
    const float omb   = 1.0f - betaH;

    const size_t base = (size_t)bh * 4096 * 64;
    const float* Qb = Q + base;
    const float* Kb = K + base;
    const float* Vb = V + base;

    // issue one segment's q/k/v tiles into buffer `bf` (12 async ops/thread)
    auto issueSeg = [&](int bf, int s) {
        const float* qg = Qb + s * 4096;
        const float* kg = Kb + s * 4096;
        const float* vg = Vb + s * 4096;
        for (int i = tid; i < 1024; i += 256) {
            const int row = i >> 4, c = (i & 15) * 4;
            async_copy16(&qg[i * 4], &q2[bf][row][c]);
            async_copy16(&kg[i * 4], &k2[bf][row][c]);
            async_copy16(&vg[i * 4], &v2[bf][row][c]);
        }
    };

    for (int i = tid; i < 4096; i += 256) Ml[i >> 6][i & 63] = 0.0f;
    if (tid < 64) zl[tid] = 0.0f;
    issueSeg(0, 0);
    __syncthreads();

    int buf = 0;
    for (int s = 0; s < 64; ++s) {
        const bool has_next = (s + 1) < 64;
        if (has_next) issueSeg(buf ^ 1, s + 1);
        wait_async_le(has_next ? 12 : 0);
        __syncthreads();

        float (*qT)[68] = q2[buf];
        float (*kT)[68] = k2[buf];
        float (*vT)[68] = v2[buf];

        // denom_q[i] = sum_d sigma(q[i,d]) * z[d] + eps
        if (tid < 64) {
            float sum = 0.0f;
            for (int d = 0; d < 64; ++d) sum += sigma_elu1(qT[tid][d]) * zl[d];
            dn[tid] = sum + 1e-6f;
        }
        __syncthreads();

        // T1 = sigma(q) @ M  (memory retrieval, unnormalized)
        gemm64<false, false, true, false>(qT, Ml, T1, 1.0f, tid);
        __syncthreads();

        // T1 = beta * T1 / denom_q ; concurrently T2 = q @ k^T / sqrt(dk)
        for (int i = tid; i < 4096; i += 256) {
            const int rr = i >> 6, cc = i & 63;
            T1[rr][cc] = betaH * T1[rr][cc] / dn[rr];
        }
        gemm64<false, true, false, false>(qT, kT, T2, 0.125f, tid);
        __syncthreads();

        // causal softmax, one row per thread
        if (tid < 64) {
            float mx = -3.402823466e38f;
            for (int j = 0; j <= tid; ++j) mx = fmaxf(mx, T2[tid][j]);
            float ssum = 0.0f;
            for (int j = 0; j < 64; ++j) {
                const float e = (j <= tid) ? __expf(T2[tid][j] - mx) : 0.0f;
                T2[tid][j] = e;
                ssum += e;
            }
            const float inv = 1.0f / ssum;
            for (int j = 0; j < 64; ++j) T2[tid][j] *= inv;
        }
        __syncthreads();

        // T1 += (1-beta) * P @ v  -> A_s
        gemm64<false, false, false, true>(T2, vT, T1, omb, tid);
        __syncthreads();

        // scatter A_s into (B,T,D) layout for the output projection
        for (int i = tid; i < 4096; i += 256) {
            const int rr = i >> 6, cc = i & 63;
            Aout[((size_t)b * 4096 + s * 64 + rr) * 1024 + h * 64 + cc] = T1[rr][cc];
        }
        // denom_k
        if (tid < 64) {
            float sum = 0.0f;
            for (int d = 0; d < 64; ++d) sum += sigma_elu1(kT[tid][d]) * zl[d];
            dn[tid] = sum + 1e-6f;
        }
        __syncthreads();

        // qT (scratch now) = sigma(k) @ M  (V_prev, unnormalized)
        gemm64<false, false, true, false>(kT, Ml, qT, 1.0f, tid);
        __syncthreads();

        // qT = v - V_prev/denom_k
        for (int i = tid; i < 4096; i += 256) {
            const int rr = i >> 6, cc = i & 63;
            qT[rr][cc] = vT[rr][cc] - qT[rr][cc] / dn[rr];
        }
        __syncthreads();

        // M += sigma(k)^T @ (v - V_prev) ; z += colsum(sigma(k))
        gemm64<true, false, true, true>(kT, qT, Ml, 1.0f, tid);
        if (tid < 64) {
            float sum = 0.0f;
            for (int t = 0; t < 64; ++t) sum += sigma_elu1(kT[t][tid]);
            zl[tid] += sum;
        }
        __syncthreads();   // protects buf reuse on the next iteration
        buf ^= 1;
    }

    for (int i = tid; i < 4096; i += 256) Mws[(size_t)bh * 4096 + i] = Ml[i >> 6][i & 63];
    if (tid < 64) zws[bh * 64 + tid] = zl[tid];
}

// ---------------------------------------------------------------------------
// Batch means of M and z into the output tail.
// ---------------------------------------------------------------------------
__global__ __launch_bounds__(256) void mean_kernel(const float* __restrict__ Mws,
                                                   const float* __restrict__ zws,
                                                   float* __restrict__ outM,
                                                   float* __restrict__ outz) {
    const int idx = blockIdx.x * blockDim.x + threadIdx.x;
    if (idx < 65536) {
        const int h = idx >> 12, i = idx & 4095;
        float s = 0.0f;
        for (int b = 0; b < 4; ++b) s += Mws[((size_t)(b * 16 + h) << 12) + i];
        outM[idx] = s * 0.25f;
    }
    if (idx < 1024) {
        const int h = idx >> 6, d = idx & 63;
        float s = 0.0f;
        for (int b = 0; b < 4; ++b) s += zws[(size_t)(b * 16 + h) * 64 + d];
        outz[idx] = s * 0.25f;
    }
}

extern "C" void kernel_launch(void* const* d_in, const int* in_sizes, int n_in,
                              void* d_out, int out_size, void* d_ws, size_t ws_size,
                              hipStream_t stream) {
    const float* x    = (const float*)d_in[0];
    const float* Wq   = (const float*)d_in[1];
    const float* Wk   = (const float*)d_in[2];
    const float* Wv   = (const float*)d_in[3];
    const float* Wo   = (const float*)d_in[4];
    const float* beta = (const float*)d_in[5];
    float* out = (float*)d_out;

    const size_t NQ = (size_t)4 * 16 * 4096 * 64;  // 16,777,216 floats per tensor
    float* Qw = (float*)d_ws;
    float* Kw = Qw + NQ;
    float* Vw = Kw + NQ;
    float* Aw = Vw + NQ;
    float* Mw = Aw + NQ;          // 64 * 4096 floats
    float* zw = Mw + 64 * 4096;   // 64 * 64 floats

    const dim3 gg(8, 128), bb(256);
    hipLaunchKernelGGL(gemm_xwT, gg, bb, 0, stream, x, Wq, Qw, 16384, 1024, 1024, 1);
    hipLaunchKernelGGL(gemm_xwT, gg, bb, 0, stream, x, Wk, Kw, 16384, 1024, 1024, 1);
    hipLaunchKernelGGL(gemm_xwT, gg, bb, 0, stream, x, Wv, Vw, 16384, 1024, 1024, 1);

    hipLaunchKernelGGL(infini_attn, dim3(64), dim3(256), 0, stream,
                       Qw, Kw, Vw, beta, Aw, Mw, zw);

    hipLaunchKernelGGL(gemm_xwT, gg, bb, 0, stream, Aw, Wo, out, 16384, 1024, 1024, 0);

    hipLaunchKernelGGL(mean_kernel, dim3(256), dim3(256), 0, stream,
                       Mw, zw, out + NQ, out + NQ + 65536);
}